// HTNE_32083405701144
// MI455X (gfx1250) — compile-verified
//
#include <hip/hip_runtime.h>

#define EMBED 128
#define HIST 5
#define NEG 2
#define WAVES 4                      // waves (batch elements) per block
#define ROW_STRIDE 132               // floats: 528B rows -> conflict-free ds_load_b64
#define STAGE (16 * ROW_STRIDE)      // 16 rows staged per wave (rows 9..15 zero)

typedef __attribute__((ext_vector_type(2))) float v2f;
typedef __attribute__((ext_vector_type(8))) float v8f;

__device__ __forceinline__ float log_sigmoid(float z) {
    // = -softplus(-z), numerically stable
    return fminf(z, 0.0f) - log1pf(expf(-fabsf(z)));
}

// Direct global -> LDS async copy (CDNA5): 16B per lane, tracked by ASYNCcnt.
// lds_off is the workgroup-relative LDS byte address (low 32 bits of generic ptr).
__device__ __forceinline__ void async_copy_b128(unsigned lds_off, const void* gptr) {
    asm volatile("global_load_async_to_lds_b128 %0, %1, off"
                 :: "v"(lds_off), "v"((unsigned long long)(uintptr_t)gptr)
                 : "memory");
}

__device__ __forceinline__ void wait_async0(void) {
    asm volatile("s_wait_asynccnt 0x0" ::: "memory");
}
__device__ __forceinline__ void wait_ds0(void) {
    asm volatile("s_wait_dscnt 0x0" ::: "memory");
}

__global__ __launch_bounds__(WAVES * 32) void htne_loss_kernel(
    const int*   __restrict__ xs,
    const int*   __restrict__ ys,
    const float* __restrict__ e_times,
    const int*   __restrict__ hs,
    const float* __restrict__ h_times,
    const int*   __restrict__ neg_node,
    const float* __restrict__ h_mask,
    const float* __restrict__ emb,
    const float* __restrict__ delta_tab,
    float*       __restrict__ out,
    int Bn)
{
    __shared__ float stage[WAVES * STAGE];       // V rows, padded stride
    __shared__ float gram[WAVES * 256];          // 16x16 Gram per wave

    const int lane = threadIdx.x & 31;
    const int wave = __builtin_amdgcn_readfirstlane((int)(threadIdx.x >> 5));
    int b = blockIdx.x * WAVES + wave;
    const bool valid = (b < Bn);
    if (!valid) b = 0;                            // clamp; full wave still executes uniformly

    float* buf = &stage[wave * STAGE];
    float* G   = &gram[wave * 256];

    // ---- gather the 9 node indices (wave-uniform) ----
    int idx[9];
    idx[0] = xs[b];
    idx[1] = ys[b];
#pragma unroll
    for (int j = 0; j < HIST; ++j) idx[2 + j] = hs[b * HIST + j];
#pragma unroll
    for (int k = 0; k < NEG; ++k)  idx[7 + k] = neg_node[b * NEG + k];

    // ---- zero rows 9..15 so unused Gram entries are clean (DS, in-order per wave)
#pragma unroll
    for (int r = 9; r < 16; ++r) {
        *(float4*)&buf[r * ROW_STRIDE + lane * 4] = make_float4(0.f, 0.f, 0.f, 0.f);
    }
    // ---- stage 9 embedding rows straight into LDS: 9 async 512B row gathers ----
#pragma unroll
    for (int r = 0; r < 9; ++r) {
        const float* src = emb + (size_t)idx[r] * EMBED + lane * 4;
        const unsigned dst = (unsigned)(uintptr_t)&buf[r * ROW_STRIDE + lane * 4];
        async_copy_b128(dst, src);
    }
    wait_async0();   // all rows resident in this wave's LDS slice (wave-local, no s_barrier)

    // ---- Gram = V * V^T via 32 chained V_WMMA_F32_16X16X4_F32 ----
    // f32 A 16x4 layout: lane(0-15)=row M, lane-half picks K{0,1}|{2,3}, VGPR picks K parity.
    // B 4x16 mirrors it with lane=N, so for a symmetric Gram the SAME fragment is A and B.
    v8f acc = {};
    const float* lanebase = &buf[(lane & 15) * ROW_STRIDE + ((lane >> 4) << 1)];
#pragma unroll
    for (int t = 0; t < EMBED / 4; ++t) {
        v2f a = *(const v2f*)(lanebase + 4 * t);  // 8B-aligned ds_load_b64 (2addr-fusable)
        acc = __builtin_amdgcn_wmma_f32_16x16x4_f32(
            /*neg_a=*/false, a, /*neg_b=*/false, a,
            /*c_mod=*/(short)0, acc, /*reuse_a=*/false, /*reuse_b=*/false);
    }

    // ---- spill Gram to LDS: VGPR i, lanes 0-15 -> M=i; lanes 16-31 -> M=8+i; N=lane&15
    {
        const int gn = lane & 15;
        const int gm = (lane >> 4) * 8;
#pragma unroll
        for (int i = 0; i < 8; ++i) G[(gm + i) * 16 + gn] = acc[i];
    }
    wait_ds0();      // wave-local: DS ops are in-order wave-wide, cross-lane reads now safe

    // ---- scalar epilogue (tiny): redundantly on all lanes, lane 0 stores ----
    // index map: 0=x, 1=y, 2..6=h0..h4, 7=n0, 8=n1 ; dist2(i,j) = Gii + Gjj - 2Gij
    const float g00 = G[0];
    const float p_mu = -(g00 + G[1 * 16 + 1] - 2.f * G[1]);

    const float et  = e_times[b];
    const float dlt = delta_tab[idx[0]];

    float alpha[HIST], dcy[HIST];
    float amax = -3.402823466e38f;
#pragma unroll
    for (int j = 0; j < HIST; ++j) {
        const int r = 2 + j;
        alpha[j] = -(g00 + G[r * 16 + r] - 2.f * G[r]);
        amax = fmaxf(amax, alpha[j]);
        const float dt = fabsf(et - h_times[b * HIST + j]);
        dcy[j] = expf(dlt * dt) * h_mask[b * HIST + j];
    }
    float attn[HIST], esum = 0.f;
#pragma unroll
    for (int j = 0; j < HIST; ++j) { attn[j] = expf(alpha[j] - amax); esum += attn[j]; }
    const float inv = 1.f / esum;

    float p_lambda = p_mu;
#pragma unroll
    for (int j = 0; j < HIST; ++j) p_lambda += attn[j] * inv * alpha[j] * dcy[j];

    float neg_ls = 0.f;
#pragma unroll
    for (int k = 0; k < NEG; ++k) {
        const int rk = 7 + k;
        const float gkk = G[rk * 16 + rk];
        float nl = -(g00 + gkk - 2.f * G[rk]);      // n_mu
#pragma unroll
        for (int j = 0; j < HIST; ++j) {
            const int r = 2 + j;
            const float na = -(G[r * 16 + r] + gkk - 2.f * G[r * 16 + rk]);
            nl += attn[j] * inv * na * dcy[j];
        }
        neg_ls += log_sigmoid(nl);
    }

    const float loss = log_sigmoid(p_lambda) - neg_ls;
    if (valid && lane == 0) out[b] = loss;
}

extern "C" void kernel_launch(void* const* d_in, const int* in_sizes, int n_in,
                              void* d_out, int out_size, void* d_ws, size_t ws_size,
                              hipStream_t stream) {
    const int*   xs        = (const int*)  d_in[0];
    const int*   ys        = (const int*)  d_in[1];
    const float* e_times   = (const float*)d_in[2];
    const int*   hs        = (const int*)  d_in[3];
    const float* h_times   = (const float*)d_in[4];
    const int*   neg_node  = (const int*)  d_in[5];
    const float* h_mask    = (const float*)d_in[6];
    const float* emb       = (const float*)d_in[7];
    const float* delta_tab = (const float*)d_in[8];
    float* out = (float*)d_out;

    const int Bn = in_sizes[0];                       // 65536
    const int blocks = (Bn + WAVES - 1) / WAVES;      // 16384 blocks of 128 threads
    htne_loss_kernel<<<blocks, WAVES * 32, 0, stream>>>(
        xs, ys, e_times, hs, h_times, neg_node, h_mask, emb, delta_tab, out, Bn);
}